// RNNPredictor_69157563400498
// MI455X (gfx1250) — compile-verified
//
#include <hip/hip_runtime.h>
#include <hip/hip_bf16.h>
#include <math.h>

#define RNN_H   64
#define RNN_T   512
#define RNN_I   4
#define RNN_O   40
#define RNN_B   4096
#define TILE_M  16
#define LDS_STRIDE 68   // 64 + 4 floats pad -> 17-bank step, conflict-free A loads
#define WAVES_PER_BLOCK 4

typedef __attribute__((ext_vector_type(2))) float v2f;
typedef __attribute__((ext_vector_type(8))) float v8f;

#if __has_builtin(__builtin_amdgcn_tanhf)
#define FAST_TANH(x) __builtin_amdgcn_tanhf(x)
#else
#define FAST_TANH(x) tanhf(x)
#endif

__device__ __forceinline__ v8f wmma_f32_16x16x4(v2f a, v2f b, v8f c) {
    return __builtin_amdgcn_wmma_f32_16x16x4_f32(
        /*neg_a=*/false, a, /*neg_b=*/false, b,
        /*c_mod=*/(short)0, c, /*reuse_a=*/false, /*reuse_b=*/false);
}

__global__ __launch_bounds__(WAVES_PER_BLOCK * 32, 1)
void rnn_fused_kernel(const float* __restrict__ x,
                      const float* __restrict__ Wih,
                      const float* __restrict__ Whh,
                      const float* __restrict__ bih,
                      const float* __restrict__ bhh,
                      const float* __restrict__ fcW,
                      const float* __restrict__ fcb,
                      float* __restrict__ out)
{
    __shared__ float hbuf[WAVES_PER_BLOCK][TILE_M * LDS_STRIDE];

    const int lane  = threadIdx.x & 31;
    const int wave  = threadIdx.x >> 5;
    const int tile  = blockIdx.x * WAVES_PER_BLOCK + wave;  // 0..255
    const int bBase = tile * TILE_M;
    const int c  = lane & 15;   // N / batch-row selector within 16
    const int hi = lane >> 4;   // half-wave selector (K split)

    float* hl = hbuf[wave];

    // ---- Preload W_hh as WMMA B-fragments (held in VGPRs across all 512 steps)
    // B[v][lane] for (nt,kc) = Whh[nt*16 + c][kc*4 + 2*hi + v]   (B = Whh^T)
    v2f whhB[4][16];
#pragma unroll
    for (int nt = 0; nt < 4; ++nt) {
        const float* rowp = Whh + (nt * 16 + c) * RNN_H;
#pragma unroll
        for (int kc = 0; kc < 16; ++kc)
            whhB[nt][kc] = *(const v2f*)(rowp + kc * 4 + 2 * hi);
    }
    // ---- W_ih^T B-fragments (K = input index, 0..3)
    v2f wihB[4];
#pragma unroll
    for (int nt = 0; nt < 4; ++nt)
        wihB[nt] = *(const v2f*)(Wih + (nt * 16 + c) * RNN_I + 2 * hi);
    // ---- Fused bias per N-tile (same value for all 8 C rows of a lane)
    float bsum[4];
#pragma unroll
    for (int nt = 0; nt < 4; ++nt)
        bsum[nt] = bih[nt * 16 + c] + bhh[nt * 16 + c];

    // ---- h0 = 0 in LDS (row-major [batch_row][hidden], stride LDS_STRIDE)
    for (int i = lane; i < TILE_M * LDS_STRIDE; i += 32) hl[i] = 0.0f;

    // Per-lane x pointer: lane loads x[bBase+c][t][2*hi .. 2*hi+1]
    const float* xptr = x + (size_t)(bBase + c) * RNN_T * RNN_I + 2 * hi;

    for (int t = 0; t < RNN_T; ++t) {
        // x_t A-fragment: A[v][lane] = x[b=c][t][K=2*hi+v]  (16x4 f32 A shape)
        v2f xA = *(const v2f*)(xptr + t * RNN_I);
        __builtin_prefetch(xptr + (t + 8) * RNN_I, 0, 0);

        // h A-fragments: A[v][lane] for kc = h[c][kc*4 + 2*hi + v]
        v2f hA[16];
#pragma unroll
        for (int kc = 0; kc < 16; ++kc)
            hA[kc] = *(const v2f*)(hl + c * LDS_STRIDE + kc * 4 + 2 * hi);

#pragma unroll
        for (int nt = 0; nt < 4; ++nt) {
            float b0 = bsum[nt];
            v8f acc = { b0, b0, b0, b0, b0, b0, b0, b0 };
            // input projection: x_t(16x4) @ Wih^T(4x16-tile)
            acc = wmma_f32_16x16x4(xA, wihB[nt], acc);
            // recurrence: h(16x64) @ Whh^T(64x16-tile), K in chunks of 4
#pragma unroll
            for (int kc = 0; kc < 16; ++kc)
                acc = wmma_f32_16x16x4(hA[kc], whhB[nt][kc], acc);
            // tanh + scatter back to LDS in C/D layout:
            // D[v][lane] = h[M = v + 8*hi][N = nt*16 + c]
#pragma unroll
            for (int v = 0; v < 8; ++v) {
                float hv = FAST_TANH(acc[v]);
                hl[(v + 8 * hi) * LDS_STRIDE + nt * 16 + c] = hv;
            }
        }
    }

    // ---- Epilogue FC: out[bBase+r][o] = h_last[r] . fcW[o] + fcb[o]
    // lane -> batch row r = c; outputs o = hi + 2*j, j = 0..19  (covers 0..39)
    const int r = c;
#pragma unroll 4
    for (int j = 0; j < 20; ++j) {
        const int o = hi + 2 * j;
        float acc = fcb[o];
        const float* wr = fcW + o * RNN_H;
#pragma unroll
        for (int h = 0; h < RNN_H; ++h)
            acc = fmaf(hl[r * LDS_STRIDE + h], wr[h], acc);
        out[(size_t)(bBase + r) * RNN_O + o] = acc;
    }
}

extern "C" void kernel_launch(void* const* d_in, const int* in_sizes, int n_in,
                              void* d_out, int out_size, void* d_ws, size_t ws_size,
                              hipStream_t stream) {
    const float* x    = (const float*)d_in[0];
    const float* Wih  = (const float*)d_in[1];
    const float* Whh  = (const float*)d_in[2];
    const float* bih  = (const float*)d_in[3];
    const float* bhh  = (const float*)d_in[4];
    const float* fcW  = (const float*)d_in[5];
    const float* fcb  = (const float*)d_in[6];
    float* out = (float*)d_out;

    const int tiles  = RNN_B / TILE_M;              // 256
    const int blocks = tiles / WAVES_PER_BLOCK;     // 64
    rnn_fused_kernel<<<blocks, WAVES_PER_BLOCK * 32, 0, stream>>>(
        x, Wih, Whh, bih, bhh, fcW, fcb, out);
}